// RotaryPositionalEmbedding_80788334838172
// MI455X (gfx1250) — compile-verified
//
#include <hip/hip_runtime.h>
#include <hip/hip_bf16.h>

typedef __attribute__((ext_vector_type(4))) float f32x4;

#define RB 8
#define RL 2048
#define RH 16
#define RD 128
#define PAIRS (RD / 2)                       // 64 (cos,sin) pairs per position
#define N_ELEM (RB * RL * RH * RD)           // 33,554,432 floats (128 MiB)
#define N_V4 (N_ELEM / 4)                    // 8,388,608 float4s
#define FLOATS_PER_BLOCK 1024                // 256 threads * float4
#define TAB_FLOATS (RL * PAIRS * 2)          // cos,sin interleaved
#define TAB_BYTES (TAB_FLOATS * sizeof(float))  // 1 MiB

// ---------------------------------------------------------------------------
// Kernel 1: build the (cos,sin) table, tab[(l*64 + j)*2] = cos, +1 = sin.
// 131072 threads, negligible runtime; keeps all transcendentals out of the
// streaming kernel. Table lives in d_ws and stays resident in the 192MB L2.
// ---------------------------------------------------------------------------
__global__ __launch_bounds__(256) void rope_table_kernel(float* __restrict__ tab) {
    int idx = blockIdx.x * blockDim.x + threadIdx.x;   // [0, RL*PAIRS)
    int p = idx >> 6;                                  // position l
    int j = idx & 63;                                  // pair index
    float expo = (float)(2 * j) * (1.0f / (float)RD);  // i/d with i = 2j
    float inv_freq = powf(10000.0f, -expo);
    float theta = (float)p * inv_freq;
    float s, c;
    sincosf(theta, &s, &c);
    tab[2 * idx + 0] = c;
    tab[2 * idx + 1] = s;
}

// Rotate one float4 (two adjacent even/odd pairs) by (c0,s0,c1,s1).
__device__ __forceinline__ f32x4 rope_rotate(f32x4 v, f32x4 cs) {
    f32x4 o;
    o.x = v.x * cs.x - v.y * cs.y;   // oe0 = xe*cos - xo*sin
    o.y = v.x * cs.y + v.y * cs.x;   // oo0 = xe*sin + xo*cos
    o.z = v.z * cs.z - v.w * cs.w;   // oe1
    o.w = v.z * cs.w + v.w * cs.z;   // oo1
    return o;
}

// ---------------------------------------------------------------------------
// Kernel 2: streaming apply. Each block handles 1024 contiguous floats =
// half of one (b,l) slice, so every thread in the block shares the same l.
// Wave 0 async-copies this l's 512-byte (cos,sin) row from L2 into LDS
// (global_load_async_to_lds_b128, ASYNCcnt), then all 8 waves broadcast-read
// it with ds_load_b128. Bulk x traffic uses non-temporal b128 so it does not
// evict the table from L2.
// ---------------------------------------------------------------------------
__global__ __launch_bounds__(256) void rope_apply_async_kernel(
    const float* __restrict__ x, float* __restrict__ out,
    const float* __restrict__ tab) {
    __shared__ __align__(16) float s_tab[2 * PAIRS];   // 512 bytes

    const int tid = threadIdx.x;
    const int l = (blockIdx.x >> 1) & (RL - 1);        // 2 blocks per l-slice

    if (tid < 32) {   // exactly wave 0 under wave32; EXEC all-ones in-branch
        // LDS flat address truncates to the wave-relative LDS byte offset.
        unsigned lds_off = (unsigned)(size_t)(&s_tab[0]) + (unsigned)tid * 16u;
        unsigned voff = (unsigned)tid * 16u;            // 32 lanes x 16B = 512B
        const float* srow = tab + (size_t)l * (2 * PAIRS);
        asm volatile("global_load_async_to_lds_b128 %0, %1, %2"
                     :: "v"(lds_off), "v"(voff), "s"(srow)
                     : "memory");
        asm volatile("s_wait_asynccnt 0x0" ::: "memory");
    }
    __syncthreads();

    // Thread t covers d-offsets [4t mod 128, +4) -> pair base j0 = 2*(t%32),
    // i.e. a 16-byte LDS read (c0,s0,c1,s1) at (t%32)*16: ds_load_b128,
    // conflict-free (lanes t and t+32 broadcast from the same address).
    f32x4 cs = ((const f32x4*)s_tab)[tid & 31];

    const long long v4 = (long long)blockIdx.x * (FLOATS_PER_BLOCK / 4) + tid;
    f32x4 v = __builtin_nontemporal_load((const f32x4*)x + v4);
    f32x4 o = rope_rotate(v, cs);
    __builtin_nontemporal_store(o, (f32x4*)out + v4);
}

// ---------------------------------------------------------------------------
// Fallback (only if d_ws can't hold the 1 MiB table): 64 lanes per block
// compute the sin/cos row in-kernel, then identical streaming path.
// ---------------------------------------------------------------------------
__global__ __launch_bounds__(256) void rope_apply_fallback_kernel(
    const float* __restrict__ x, float* __restrict__ out) {
    __shared__ __align__(16) float s_tab[2 * PAIRS];

    const int tid = threadIdx.x;
    const int l = (blockIdx.x >> 1) & (RL - 1);

    if (tid < PAIRS) {
        float expo = (float)(2 * tid) * (1.0f / (float)RD);
        float inv_freq = powf(10000.0f, -expo);
        float theta = (float)l * inv_freq;
        float s, c;
        sincosf(theta, &s, &c);
        s_tab[2 * tid + 0] = c;
        s_tab[2 * tid + 1] = s;
    }
    __syncthreads();

    f32x4 cs = ((const f32x4*)s_tab)[tid & 31];
    const long long v4 = (long long)blockIdx.x * (FLOATS_PER_BLOCK / 4) + tid;
    f32x4 v = __builtin_nontemporal_load((const f32x4*)x + v4);
    f32x4 o = rope_rotate(v, cs);
    __builtin_nontemporal_store(o, (f32x4*)out + v4);
}

extern "C" void kernel_launch(void* const* d_in, const int* in_sizes, int n_in,
                              void* d_out, int out_size, void* d_ws, size_t ws_size,
                              hipStream_t stream) {
    (void)in_sizes; (void)n_in; (void)out_size;
    const float* x = (const float*)d_in[0];
    float* out = (float*)d_out;

    const int apply_blocks = N_V4 / 256;   // 32768 blocks of 256 threads

    if (ws_size >= (size_t)TAB_BYTES) {
        float* tab = (float*)d_ws;
        rope_table_kernel<<<(RL * PAIRS) / 256, 256, 0, stream>>>(tab);
        rope_apply_async_kernel<<<apply_blocks, 256, 0, stream>>>(x, out, tab);
    } else {
        rope_apply_fallback_kernel<<<apply_blocks, 256, 0, stream>>>(x, out);
    }
}